// ProposalTargetAssigner_3143916060964
// MI455X (gfx1250) — compile-verified
//
#include <hip/hip_runtime.h>
#include <hip/hip_bf16.h>
#include <math.h>

// ---------------- problem constants ----------------
#define NB        40
#define N_CLS     1
#define N_YAW     2
#define NY        200
#define NX        176
#define NANCH     (N_YAW * NY * NX)      // 70400
#define TOT       (N_CLS * NANCH)        // 70400
#define DOF       7
#define NUM_SAMPLES 512
#define NUM_POS_MAX 256                  // int(512 * 0.5)
#define IOU_LO    0.45f
#define IOU_HI    0.6f
#define CLS_OUT   ((N_CLS + 1) * NANCH)  // 140800

static_assert(N_CLS == 1, "conflict-resolution fast path assumes one class");
static_assert(TOT % 256 == 0, "grid sizing assumes exact multiple");
static_assert((NB * 16) % 4 == 0, "boxinfo must be a whole number of 16B chunks");

typedef __attribute__((ext_vector_type(2))) float v2f;
typedef __attribute__((ext_vector_type(8))) float v8f;
typedef __attribute__((ext_vector_type(4))) int v4i;
typedef unsigned long long u64;

#if defined(__gfx1250__) && __has_builtin(__builtin_amdgcn_global_load_async_to_lds_b128) && \
    __has_builtin(__builtin_amdgcn_s_wait_asynccnt)
#define HAVE_ASYNC_LDS 1
#else
#define HAVE_ASYNC_LDS 0
#endif

// ---------------- helpers ----------------

// order-preserving map float -> uint (monotonic for all floats)
__device__ __forceinline__ unsigned ford(float f) {
  unsigned u = __float_as_uint(f);
  return (u & 0x80000000u) ? ~u : (u | 0x80000000u);
}
__device__ __forceinline__ float funord(unsigned u) {
  unsigned b = (u & 0x80000000u) ? (u & 0x7FFFFFFFu) : ~u;
  return __uint_as_float(b);
}

__device__ __forceinline__ unsigned mix32(unsigned x) {
  x ^= x >> 16; x *= 0x7FEB352Du;
  x ^= x >> 15; x *= 0x846CA68Bu;
  x ^= x >> 16; return x;
}
// unique deterministic sampling keys (random high bits, index in low 17 bits)
__device__ __forceinline__ unsigned key_pos(int i) {
  return (mix32(0x243F6A88u ^ (unsigned)i) & 0xFFFE0000u) | (unsigned)i;
}
__device__ __forceinline__ unsigned key_neg(int i) {
  return (mix32(0xB7E15162u ^ (unsigned)i) & 0xFFFE0000u) | (unsigned)i;
}

// Stage the 2.5KB box-record table into LDS. On gfx1250, use the async
// LDS-copy path (GLOBAL_LOAD_ASYNC_TO_LDS_B128 + s_wait_asynccnt, ASYNCcnt).
__device__ __forceinline__ void stage_boxinfo(const float* __restrict__ boxinfo,
                                              float* s_box, int tid, int nthreads) {
#if HAVE_ASYNC_LDS
  const int nchunks = (NB * 16) / 4;  // 160 x 16B
  for (int c = tid; c < nchunks; c += nthreads) {
    __builtin_amdgcn_global_load_async_to_lds_b128(
        (__attribute__((address_space(1))) v4i*)(size_t)(boxinfo + c * 4),
        (__attribute__((address_space(3))) v4i*)(unsigned)(size_t)(s_box + c * 4),
        0, 0);
  }
  __builtin_amdgcn_s_wait_asynccnt(0);
#else
  for (int i = tid; i < NB * 16; i += nthreads) s_box[i] = boxinfo[i];
#endif
}

struct ABox {
  float cx[4], cy[4];
  float area, xmin, xmax, ymin, ymax;
};

__device__ __attribute__((noinline)) void make_anchor(const float* ap, ABox& A) {
  float x = ap[0], y = ap[1], w = ap[3], l = ap[4], yaw = ap[6];
  float c = cosf(yaw), s = sinf(yaw);
  const float dxs[4] = {0.5f, -0.5f, -0.5f, 0.5f};
  const float dys[4] = {0.5f, 0.5f, -0.5f, -0.5f};
  float xmin = 1e30f, xmax = -1e30f, ymin = 1e30f, ymax = -1e30f;
#pragma unroll
  for (int k = 0; k < 4; k++) {
    float dx = dxs[k] * w, dy = dys[k] * l;
    float px = x + c * dx - s * dy;
    float py = y + s * dx + c * dy;
    A.cx[k] = px; A.cy[k] = py;
    xmin = fminf(xmin, px); xmax = fmaxf(xmax, px);
    ymin = fminf(ymin, py); ymax = fmaxf(ymax, py);
  }
  A.area = w * l;
  A.xmin = xmin; A.xmax = xmax; A.ymin = ymin; A.ymax = ymax;
}

// Exact rotated-rect IoU of one GT box (precomputed record sb) vs anchor A.
// noinline => bit-identical results across passes on the fallback path.
// sb layout: [0..7] corners x0,y0..x3,y3 ; [8] area ; [9..12] aabb
__device__ __attribute__((noinline)) float pair_iou(const float* sb, const ABox& A) {
  float bxmin = sb[9], bxmax = sb[10], bymin = sb[11], bymax = sb[12];
  // AABB prefilter: disjoint => intersection area exactly 0 => IoU exactly 0
  if (bxmax < A.xmin || bxmin > A.xmax || bymax < A.ymin || bymin > A.ymax)
    return 0.0f;

  float px[8], py[8];
#pragma unroll
  for (int i = 0; i < 4; i++) { px[i] = sb[2 * i]; py[i] = sb[2 * i + 1]; }
  int n = 4;

#pragma unroll
  for (int k = 0; k < 4; k++) {     // clip by anchor edges (CCW)
    float ax = A.cx[k], ay = A.cy[k];
    float bx = A.cx[(k + 1) & 3], by = A.cy[(k + 1) & 3];
    float ex = bx - ax, ey = by - ay;
    float qx[8], qy[8];
    int m = 0;
    for (int j = 0; j < 8; j++) {
      if (j >= n) break;
      int jn = (j + 1 == n) ? 0 : j + 1;
      float dp = ex * (py[j] - ay) - ey * (px[j] - ax);
      float dq = ex * (py[jn] - ay) - ey * (px[jn] - ax);
      bool qin = (dq >= 0.0f);
      if ((dp >= 0.0f) != qin) {
        float t = dp / (dp - dq);
        qx[m] = px[j] + t * (px[jn] - px[j]);
        qy[m] = py[j] + t * (py[jn] - py[j]);
        m++;
      }
      if (qin) { qx[m] = px[jn]; qy[m] = py[jn]; m++; }
    }
    n = m;
    for (int j = 0; j < 8; j++) {
      px[j] = (j < m) ? qx[j] : 0.0f;
      py[j] = (j < m) ? qy[j] : 0.0f;
    }
  }

  float s = 0.0f;
  for (int j = 0; j < 8; j++) {
    if (j >= n) break;
    int jn = (j + 1 == n) ? 0 : j + 1;
    s += px[j] * py[jn] - px[jn] * py[j];
  }
  float inter = 0.5f * fabsf(s);
  float uni = sb[8] + A.area - inter;
  return inter / fmaxf(uni, 1e-8f);
}

// ---------------- K1: GT-box corners via WMMA ----------------
// corners(40x8) = F(40x6) @ W(6x8):
//   F = [x, y, .5cw, .5sw, .5cl, .5sl], W constant {0, +-1}
// 3 M-tiles of 16, K split into two x4 chunks -> 6 v_wmma_f32_16x16x4_f32.

__device__ __forceinline__ float boxfeat(const float* boxes, int b, int k) {
  if (b >= NB || k >= 6) return 0.0f;
  const float* p = boxes + b * DOF;
  if (k == 0) return p[0];
  if (k == 1) return p[1];
  float c = cosf(p[6]), s = sinf(p[6]);
  float w = p[3], l = p[4];
  switch (k) {
    case 2: return 0.5f * c * w;
    case 3: return 0.5f * s * w;
    case 4: return 0.5f * c * l;
    default: return 0.5f * s * l;   // k == 5
  }
}

__device__ __forceinline__ float wmat(int k, int n) {
  if (n >= 8 || k >= 6) return 0.0f;
  int corner = n >> 1;
  int isY = n & 1;
  float alpha = (corner == 0 || corner == 3) ? 1.0f : -1.0f;  // [1,-1,-1,1]
  float beta  = (corner < 2) ? 1.0f : -1.0f;                  // [1,1,-1,-1]
  switch (k) {
    case 0: return isY ? 0.0f : 1.0f;   // x -> X
    case 1: return isY ? 1.0f : 0.0f;   // y -> Y
    case 2: return isY ? 0.0f : alpha;  // .5cw -> X
    case 3: return isY ? alpha : 0.0f;  // .5sw -> Y
    case 4: return isY ? beta : 0.0f;   // .5cl -> Y
    default: return isY ? 0.0f : -beta; // .5sl -> X (k==5)
  }
}

__global__ __launch_bounds__(32) void corners_kernel(const float* __restrict__ boxes,
                                                     float* __restrict__ boxinfo) {
#if __has_builtin(__builtin_amdgcn_wmma_f32_16x16x4_f32)
  const int lane = threadIdx.x;
  const int half = lane >> 4;    // K-half of the x4 fragment
  const int l16  = lane & 15;    // M (for A) / N (for B)
#pragma unroll
  for (int tile = 0; tile < 3; tile++) {
    v8f acc;
#pragma unroll
    for (int j = 0; j < 8; j++) acc[j] = 0.0f;
#pragma unroll
    for (int ck = 0; ck < 2; ck++) {
      int kbase = ck * 4 + half * 2;
      v2f a, bf;
      a.x  = boxfeat(boxes, tile * 16 + l16, kbase + 0);
      a.y  = boxfeat(boxes, tile * 16 + l16, kbase + 1);
      bf.x = wmat(kbase + 0, l16);
      bf.y = wmat(kbase + 1, l16);
      acc = __builtin_amdgcn_wmma_f32_16x16x4_f32(
          /*neg_a=*/false, a, /*neg_b=*/false, bf,
          /*c_mod=*/(short)0, acc, /*reuse_a=*/false, /*reuse_b=*/false);
    }
#pragma unroll
    for (int j = 0; j < 8; j++) {
      int m = tile * 16 + j + half * 8;   // C/D layout: vgpr j = rows j / j+8
      if (m < NB && l16 < 8) boxinfo[m * 16 + l16] = acc[j];
    }
  }
#else
  for (int b = threadIdx.x; b < NB; b += 32) {
    const float* p = boxes + b * DOF;
    float c = cosf(p[6]), s = sinf(p[6]);
    const float al[4] = {1, -1, -1, 1}, be[4] = {1, 1, -1, -1};
    for (int k = 0; k < 4; k++) {
      float dx = 0.5f * al[k] * p[3], dy = 0.5f * be[k] * p[4];
      boxinfo[b * 16 + 2 * k]     = p[0] + c * dx - s * dy;
      boxinfo[b * 16 + 2 * k + 1] = p[1] + s * dx + c * dy;
    }
  }
#endif
}

// ---------------- K2: finish box records + init reductions ----------------
__global__ __launch_bounds__(64) void boxfinish_kernel(const float* __restrict__ boxes,
                                                       const int* __restrict__ cls,
                                                       float* __restrict__ boxinfo,
                                                       unsigned* __restrict__ gt_ord,
                                                       int* __restrict__ hasf) {
  int t = threadIdx.x;
  for (int b = t; b < NB; b += 64) {
    float xmin = 1e30f, xmax = -1e30f, ymin = 1e30f, ymax = -1e30f;
    for (int k = 0; k < 4; k++) {
      float x = boxinfo[b * 16 + 2 * k];
      float y = boxinfo[b * 16 + 2 * k + 1];
      xmin = fminf(xmin, x); xmax = fmaxf(xmax, x);
      ymin = fminf(ymin, y); ymax = fmaxf(ymax, y);
    }
    boxinfo[b * 16 + 8]  = boxes[b * DOF + 3] * boxes[b * DOF + 4];  // area w*l
    boxinfo[b * 16 + 9]  = xmin;
    boxinfo[b * 16 + 10] = xmax;
    boxinfo[b * 16 + 11] = ymin;
    boxinfo[b * 16 + 12] = ymax;
    ((int*)boxinfo)[b * 16 + 13] = cls[b];
  }
  for (int i = t; i < N_CLS * NB; i += 64) gt_ord[i] = 0u;  // ord(-inf)
  if (t < N_CLS) {
    int h = 0;
    for (int b = 0; b < NB; b++) h |= (cls[b] == t);
    hasf[t] = h;
  }
}

// ---------------- K3: per-box max IoU over all anchors ----------------
// Optionally caches the full pair matrix (11.3MB -> L2-resident on MI455X)
// so the label pass becomes a pure streaming pass with exact bit-equality.
__global__ __launch_bounds__(256) void rowmax_kernel(const float* __restrict__ anchors,
                                                     const float* __restrict__ boxinfo,
                                                     unsigned* __restrict__ gt_ord,
                                                     float* __restrict__ iouM,
                                                     int store_iou) {
  __shared__ float s_box[NB * 16];
  __shared__ unsigned s_ord[N_CLS * NB];
  int tid = threadIdx.x;
  stage_boxinfo(boxinfo, s_box, tid, 256);
  for (int i = tid; i < N_CLS * NB; i += 256) s_ord[i] = 0u;
  __syncthreads();

  int idx = blockIdx.x * 256 + tid;
  if (idx < TOT) {
    int c = idx / NANCH;
    const float* ap = anchors + (size_t)idx * DOF;
    if (idx + 256 < TOT) __builtin_prefetch(anchors + (size_t)(idx + 256) * DOF, 0, 1);
    ABox A;
    make_anchor(ap, A);
    for (int b = 0; b < NB; b++) {
      int bcls = ((const int*)s_box)[b * 16 + 13];
      float pv = (bcls == c) ? pair_iou(&s_box[b * 16], A) : -1.0f;
      if (store_iou) iouM[(size_t)idx * NB + b] = pv;
      atomicMax(&s_ord[c * NB + b], ford(pv));
    }
  }
  __syncthreads();
  for (int i = tid; i < N_CLS * NB; i += 256) atomicMax(&gt_ord[i], s_ord[i]);
}

// ---------------- K4: pre-sampling labels + matches ----------------
__global__ __launch_bounds__(256) void labels_kernel(const float* __restrict__ anchors,
                                                     const float* __restrict__ boxinfo,
                                                     const unsigned* __restrict__ gt_ord,
                                                     const int* __restrict__ hasf,
                                                     int* __restrict__ labels,
                                                     int* __restrict__ matches,
                                                     const float* __restrict__ iouM,
                                                     int store_iou) {
  __shared__ float s_box[NB * 16];
  __shared__ float s_gt[N_CLS * NB];
  __shared__ int s_has[N_CLS];
  int tid = threadIdx.x;
  stage_boxinfo(boxinfo, s_box, tid, 256);
  for (int i = tid; i < N_CLS * NB; i += 256) s_gt[i] = funord(gt_ord[i]);
  if (tid < N_CLS) s_has[tid] = hasf[tid];
  __syncthreads();

  int idx = blockIdx.x * 256 + tid;
  if (idx >= TOT) return;
  int c = idx / NANCH;

  float best = -1.0f;
  int bestb = -1;
  bool lowq = false;

  if (store_iou) {
    // replay cached pair values (bit-identical to K3 by construction)
    const float* row = iouM + (size_t)idx * NB;
    for (int b = 0; b < NB; b++) {
      float pv = row[b];
      if (pv > best) { best = pv; bestb = b; }        // first-max == argmax
      float g = s_gt[c * NB + b];
      if (pv == g && g > 0.0f) lowq = true;           // low-quality match
    }
  } else {
    const float* ap = anchors + (size_t)idx * DOF;
    if (idx + 256 < TOT) __builtin_prefetch(anchors + (size_t)(idx + 256) * DOF, 0, 1);
    ABox A;
    make_anchor(ap, A);
    for (int b = 0; b < NB; b++) {
      int bcls = ((const int*)s_box)[b * 16 + 13];
      float pv = (bcls == c) ? pair_iou(&s_box[b * 16], A) : -1.0f;
      if (pv > best) { best = pv; bestb = b; }
      float g = s_gt[c * NB + b];
      if (pv == g && g > 0.0f) lowq = true;
    }
  }

  int lab = (best >= IOU_HI) ? 1 : ((best >= IOU_LO) ? -1 : 0);
  if (lowq) lab = 1;
  if (!s_has[c]) { lab = -1; bestb = -1; }
  labels[idx] = lab;
  matches[idx] = bestb;
}

// ---------------- K5: deterministic subsample thresholds ----------------
// Exact k-smallest-key selection via binary search on unique 32-bit keys.
__global__ __launch_bounds__(256) void select_kernel(const int* __restrict__ labels,
                                                     u64* __restrict__ selT) {
  __shared__ int s_cnt, s_P, s_N;
  int tid = threadIdx.x;

  int lp = 0, ln = 0;
  for (int i = tid; i < TOT; i += 256) {
    int L = labels[i];
    lp += (L == 1);
    ln += (L == 0);
  }
  if (tid == 0) { s_P = 0; s_N = 0; }
  __syncthreads();
  atomicAdd(&s_P, lp);
  atomicAdd(&s_N, ln);
  __syncthreads();
  int P = s_P, Ngt = s_N;
  int num_pos = (P < NUM_POS_MAX) ? P : NUM_POS_MAX;
  int rem = NUM_SAMPLES - num_pos;
  int num_neg = (Ngt < rem) ? Ngt : rem;

  // positives
  u64 lo = 0, hi = (1ull << 32);
  while (lo < hi) {
    u64 mid = (lo + hi) >> 1;
    __syncthreads();
    if (tid == 0) s_cnt = 0;
    __syncthreads();
    int loc = 0;
    for (int i = tid; i < TOT; i += 256)
      if (labels[i] == 1 && (u64)key_pos(i) < mid) loc++;
    atomicAdd(&s_cnt, loc);
    __syncthreads();
    if (s_cnt >= num_pos) hi = mid; else lo = mid + 1;
  }
  u64 Tpos = lo;

  // negatives
  lo = 0; hi = (1ull << 32);
  while (lo < hi) {
    u64 mid = (lo + hi) >> 1;
    __syncthreads();
    if (tid == 0) s_cnt = 0;
    __syncthreads();
    int loc = 0;
    for (int i = tid; i < TOT; i += 256)
      if (labels[i] == 0 && (u64)key_neg(i) < mid) loc++;
    atomicAdd(&s_cnt, loc);
    __syncthreads();
    if (s_cnt >= num_neg) hi = mid; else lo = mid + 1;
  }
  u64 Tneg = lo;

  if (tid == 0) { selT[0] = Tpos; selT[1] = Tneg; }
}

// ---------------- K6: finalize cls + reg targets ----------------
__global__ __launch_bounds__(256) void finalize_kernel(const float* __restrict__ boxes,
                                                       const float* __restrict__ anchors,
                                                       const int* __restrict__ labels,
                                                       const int* __restrict__ matches,
                                                       const u64* __restrict__ selT,
                                                       float* __restrict__ out) {
  int idx = blockIdx.x * 256 + threadIdx.x;
  if (idx >= TOT) return;
  u64 Tp = selT[0], Tn = selT[1];
  int lab = labels[idx];
  bool selp = (lab == 1) && ((u64)key_pos(idx) < Tp);
  bool seln = (lab == 0) && ((u64)key_neg(idx) < Tn);
  int f = selp ? 1 : (seln ? 0 : -1);

  // N_CLS == 1: no ambiguity; ignore := (f == -1); -1 mapped to 0 in labels row
  bool ignore = (f == -1);
  out[idx] = ignore ? 0.0f : (float)f;                 // labels row (class 0)
  out[N_CLS * NANCH + (idx % NANCH)] = ignore ? 1.0f : 0.0f;  // ignore row

  int m = matches[idx];
  int safe = (m < 0) ? 0 : ((m > NB - 1) ? NB - 1 : m);
  const float* g = boxes + safe * DOF;
  const float* an = anchors + (size_t)idx * DOF;
  float r[7];
  r[0] = g[0] - an[0];
  r[1] = g[1] - an[1];
  r[2] = g[2] - an[2];
  r[3] = (g[3] - an[3]) / an[3];
  r[4] = (g[4] - an[4]) / an[4];
  r[5] = (g[5] - an[5]) / an[5];
  r[6] = g[6] - an[6];
  float* ro = out + CLS_OUT + (size_t)idx * DOF;
  bool pos = (f == 1);
#pragma unroll
  for (int k = 0; k < 7; k++) ro[k] = pos ? r[k] : 0.0f;
}

// ---------------- host launcher ----------------
extern "C" void kernel_launch(void* const* d_in, const int* in_sizes, int n_in,
                              void* d_out, int out_size, void* d_ws, size_t ws_size,
                              hipStream_t stream) {
  const float* boxes  = (const float*)d_in[0];   // (40, 7)
  const int* cls      = (const int*)d_in[1];     // (40,)
  const float* anchors = (const float*)d_in[2];  // (1, 2, 200, 176, 7)
  float* out = (float*)d_out;                    // 140800 cls + 492800 reg

  char* ws = (char*)d_ws;
  float* boxinfo    = (float*)ws;                      // 40 * 16 floats
  unsigned* gt_ord  = (unsigned*)(ws + 4096);          // N_CLS * 40
  int* hasf         = (int*)(ws + 4096 + 512);
  u64* selT         = (u64*)(ws + 4096 + 1024);
  int* labels       = (int*)(ws + 8192);               // TOT ints
  int* matches      = labels + TOT;                    // TOT ints
  const size_t iou_off = 8192 + 2 * (size_t)TOT * sizeof(int);
  float* iouM       = (float*)(ws + iou_off);          // TOT * NB floats (optional)
  int store_iou = (ws_size >= iou_off + (size_t)TOT * NB * sizeof(float)) ? 1 : 0;

  corners_kernel<<<1, 32, 0, stream>>>(boxes, boxinfo);
  boxfinish_kernel<<<1, 64, 0, stream>>>(boxes, cls, boxinfo, gt_ord, hasf);
  rowmax_kernel<<<TOT / 256, 256, 0, stream>>>(anchors, boxinfo, gt_ord, iouM, store_iou);
  labels_kernel<<<TOT / 256, 256, 0, stream>>>(anchors, boxinfo, gt_ord, hasf,
                                               labels, matches, iouM, store_iou);
  select_kernel<<<1, 256, 0, stream>>>(labels, selT);
  finalize_kernel<<<TOT / 256, 256, 0, stream>>>(boxes, anchors, labels, matches,
                                                 selT, out);
}